// MultiHeadedAttention_2_18631568130124
// MI455X (gfx1250) — compile-verified
//
#include <hip/hip_runtime.h>
#include <hip/hip_bf16.h>

typedef float v2f __attribute__((ext_vector_type(2)));
typedef float v8f __attribute__((ext_vector_type(8)));

#define B_  16
#define H_  16
#define DK  64
#define LK  8192
#define DM  1024          // H_*DK
#define TILE 16
#define NTILES (LK / TILE)          // 512
#define SPLIT 4
#define WAVES_PER_WG 8
#define WAVES_PER_BH (SPLIT * WAVES_PER_WG)  // 32 waves share one (b,h)

// One wave computes 16-key tiles of cosine-sim scores for one (b,h) pair.
// WMMA f32 16x16x4: A = q broadcast into all 16 M-rows, B = 16 key rows
// along N. After 16 chained WMMAs (K=64), c[0] in lane n == score of key
// (l0 + (n&15)) in BOTH half-waves -> no divergence in the epilogue.
__global__ __launch_bounds__(256) void cosattn_wmma_kernel(
    const float* __restrict__ query,   // [B, 1, DM]
    const float* __restrict__ key,     // [B, LK, DM]
    const int*   __restrict__ mask,    // [B, LK]
    float* __restrict__ out)           // [B, H, LK]
{
    const int lane  = threadIdx.x & 31;
    const int wave  = threadIdx.x >> 5;
    const int bh    = blockIdx.x / SPLIT;
    const int split = blockIdx.x % SPLIT;
    const int b     = bh / H_;
    const int h     = bh % H_;
    const int waveId = split * WAVES_PER_WG + wave;   // 0..31 within (b,h)

    const int half = lane >> 4;   // which half-wave (selects K pair 0/1 vs 2/3)
    const int r    = lane & 15;   // M/N index within the 16x16 tile

    // ---- preload A operand: q replicated over all 16 rows (32 VGPRs) ----
    // A layout (f32 16x4): v0 = K={0|2}, v1 = K={1|3}, split by half-wave.
    const float* qptr = query + (size_t)b * DM + h * DK + half * 2;
    v2f qa[16];
#pragma unroll
    for (int s = 0; s < 16; ++s)
        qa[s] = *(const v2f*)(qptr + 4 * s);

    // ||q|| for this (b,h): each half-wave holds half the features.
    float qs = 0.f;
#pragma unroll
    for (int s = 0; s < 16; ++s)
        qs += qa[s].x * qa[s].x + qa[s].y * qa[s].y;
    qs += __shfl_xor(qs, 16, 32);
    const float qn = sqrtf(qs);

    const float* kbase = key  + (size_t)b * LK * DM + h * DK + half * 2;
    const int*   mbase = mask + (size_t)b * LK;
    float*       obase = out  + ((size_t)b * H_ + h) * LK;

    for (int t = waveId; t < NTILES; t += WAVES_PER_BH) {
        const int l0 = t * TILE;
        const float* rowptr = kbase + (size_t)(l0 + r) * DM;

        // Prefetch this wave's next tile. Branch-free: clamp the tile index
        // so the address is always in-bounds and EXEC never toggles.
        {
            int tpf = t + WAVES_PER_BH;
            tpf = (tpf < NTILES) ? tpf : (NTILES - 1);
            const float* pf = kbase + (size_t)(tpf * TILE + r) * DM + half * 30;
            __builtin_prefetch(pf, 0, 3);   // near-scope, covers both 128B lines
        }

        v8f  c  = {};
        float ks = 0.f;   // sum of squares of key row r (this half's features)
#pragma unroll
        for (int s = 0; s < 16; ++s) {
            // B layout (f32 4x16): v0 = row K={0|2}, v1 = row K={1|3}, N = lane.
            v2f bm = *(const v2f*)(rowptr + 4 * s);
            ks += bm.x * bm.x + bm.y * bm.y;
            c = __builtin_amdgcn_wmma_f32_16x16x4_f32(
                    /*neg_a=*/false, qa[s], /*neg_b=*/false, bm,
                    /*c_mod=*/(short)0, c, /*reuse_a=*/false, /*reuse_b=*/false);
        }
        ks += __shfl_xor(ks, 16, 32);
        const float kn    = sqrtf(ks);
        const float den   = fmaxf(qn * kn, 1e-8f);
        const float score = c[0] / den;              // identical in both halves
        const int   m     = mbase[l0 + r];
        // relu(where(mask==0, -1e9, score)) / Lk  ==  mask ? max(score,0)/Lk : 0
        const float p = (m != 0) ? fmaxf(score, 0.f) * (1.0f / LK) : 0.f;
        // Lanes r and r+16 write the same value to the same address: benign,
        // keeps EXEC fully enabled (WMMA requires EXEC == all ones).
        obase[l0 + r] = p;
    }
}

extern "C" void kernel_launch(void* const* d_in, const int* in_sizes, int n_in,
                              void* d_out, int out_size, void* d_ws, size_t ws_size,
                              hipStream_t stream) {
    (void)in_sizes; (void)n_in; (void)out_size; (void)d_ws; (void)ws_size;
    const float* query = (const float*)d_in[0];
    const float* key   = (const float*)d_in[1];
    const int*   mask  = (const int*)d_in[2];
    float* out = (float*)d_out;

    dim3 grid(B_ * H_ * SPLIT);   // 1024 workgroups
    dim3 block(256);              // 8 waves each
    hipLaunchKernelGGL(cosattn_wmma_kernel, grid, block, 0, stream,
                       query, key, mask, out);
}